// ResidualRNN_24386824307244
// MI455X (gfx1250) — compile-verified
//
#include <hip/hip_runtime.h>
#include <math.h>

// Problem constants (from reference)
#define BB 32
#define TT 512
#define DMODEL 768
#define DH 64
#define DMLP 128
#define DZ 64
#define NCAT 8
#define NCLS 8

typedef __attribute__((ext_vector_type(2))) float v2f;
typedef __attribute__((ext_vector_type(8))) float v8f;

// ---------------------------------------------------------------------------
// WMMA helpers: V_WMMA_F32_16X16X4_F32  (D = A(16x4) * B(4x16) + C, fp32)
// A fragment: lanes 0-15 -> M=lane, VGPR0=K0,VGPR1=K1 ; lanes 16-31 -> K2,K3
// B fragment: lanes 0-15 -> N=lane,  VGPR0=K0,VGPR1=K1 ; lanes 16-31 -> K2,K3
// C/D: VGPR r -> M = r + (lane>=16)*8, N = lane&15
// ---------------------------------------------------------------------------
__device__ __forceinline__ v8f wmma4(v2f a, v2f b, v8f c) {
  return __builtin_amdgcn_wmma_f32_16x16x4_f32(
      /*neg_a=*/false, a, /*neg_b=*/false, b,
      /*c_mod=*/(short)0, c, /*reuse_a=*/false, /*reuse_b=*/false);
}

// Tile GEMM: accumulate A(16xK) @ B(Kx16) into c.
// A: pointer to A[m0][0], row-major, stride lda   (rows = M)
// B (BT=false): pointer to B[0][n0], row-major, stride ldb (rows = K)
// B (BT=true ): pointer to Bt[n0][0], row-major, stride ldb (rows = N) i.e. B[k][n]=Bt[n][k]
template <bool BT>
__device__ __forceinline__ v8f tileGemm(const float* __restrict__ A, int lda,
                                        const float* __restrict__ B, int ldb,
                                        int K, v8f c) {
  const int lane = threadIdx.x & 31;
  const int l = lane & 15;
  const int kh = (lane >> 4) * 2;
  for (int k0 = 0; k0 < K; k0 += 4) {
    v2f a, b;
    a.x = A[l * lda + k0 + kh];
    a.y = A[l * lda + k0 + kh + 1];
    if (BT) {
      b.x = B[l * ldb + k0 + kh];
      b.y = B[l * ldb + k0 + kh + 1];
    } else {
      b.x = B[(k0 + kh) * ldb + l];
      b.y = B[(k0 + kh + 1) * ldb + l];
    }
    c = wmma4(a, b, c);
  }
  return c;
}

__device__ __forceinline__ v8f loadC(const float* __restrict__ C, int ldc) {
  const int lane = threadIdx.x & 31;
  const int col = lane & 15;
  const int rb = (lane >> 4) * 8;
  v8f c;
  for (int r = 0; r < 8; ++r) c[r] = C[(rb + r) * ldc + col];
  return c;
}

__device__ __forceinline__ void storeC(float* __restrict__ C, int ldc, v8f d) {
  const int lane = threadIdx.x & 31;
  const int col = lane & 15;
  const int rb = (lane >> 4) * 8;
  for (int r = 0; r < 8; ++r) C[(rb + r) * ldc + col] = d[r];
}

__device__ __forceinline__ v8f biasC(const float* __restrict__ bias, int n0) {
  const float bv = bias[n0 + (threadIdx.x & 15)];
  v8f c;
  for (int r = 0; r < 8; ++r) c[r] = bv;
  return c;
}

__device__ __forceinline__ float gelu_erf(float x) {
  return 0.5f * x * (1.0f + erff(x * 0.70710678118654752f));
}
__device__ __forceinline__ v8f geluC(v8f c) {
  for (int r = 0; r < 8; ++r) c[r] = gelu_erf(c[r]);
  return c;
}
__device__ __forceinline__ float sigmoidf(float x) {
  return 1.0f / (1.0f + expf(-x));
}

// ---------------------------------------------------------------------------
// Phase 1 (parallel): U = r1_seq(16384x768) @ lat_W1[:768,:](768x128)
// one 16x16 tile per wave; 1024 blocks x 8 waves = 8192 tiles
// ---------------------------------------------------------------------------
__global__ __launch_bounds__(256) void phase1_kernel(
    const float* __restrict__ r1, const float* __restrict__ latW1,
    float* __restrict__ U) {
  const int gid = blockIdx.x * 8 + (threadIdx.x >> 5);
  const int mt = gid >> 3, nt = gid & 7;
  const int m0 = mt * 16, n0 = nt * 16;
  v8f c = {};
  c = tileGemm<false>(r1 + (size_t)m0 * DMODEL, DMODEL, latW1 + n0, DMLP,
                      DMODEL, c);
  storeC(U + (size_t)m0 * DMLP + n0, DMLP, c);
}

// ---------------------------------------------------------------------------
// Phase 3 (parallel): r2 = a_dec(16384x128) @ dec_W2(128x768) + dec_b2
// 6144 blocks x 8 waves = 49152 tiles (1024 M-tiles x 48 N-tiles)
// ---------------------------------------------------------------------------
__global__ __launch_bounds__(256) void phase3_kernel(
    const float* __restrict__ a_dec, const float* __restrict__ decW2,
    const float* __restrict__ decB2, float* __restrict__ r2) {
  const int gid = blockIdx.x * 8 + (threadIdx.x >> 5);
  const int mt = gid / 48, nt = gid % 48;
  const int m0 = mt * 16, n0 = nt * 16;
  v8f c = biasC(decB2, n0);
  c = tileGemm<false>(a_dec + (size_t)m0 * DMLP, DMLP, decW2 + n0, DMODEL,
                      DMLP, c);
  storeC(r2 + (size_t)m0 * DMODEL + n0, DMODEL, c);
}

// ---------------------------------------------------------------------------
// Phase 2 (sequential): one persistent workgroup, 8 wave32s, T=512 steps.
// ---------------------------------------------------------------------------
__global__ __launch_bounds__(256) void seq_kernel(
    const float* __restrict__ U,        // (B*T, 128): row b*T+t
    const float* __restrict__ latW1,    // (832,128) -> h part at rows 768..831
    const float* __restrict__ latB1,    // (128)
    const float* __restrict__ latW2,    // (128,64)
    const float* __restrict__ latB2,    // (64)
    const float* __restrict__ Wih,      // (192,64)
    const float* __restrict__ bih,      // (192)
    const float* __restrict__ Whh,      // (192,64)
    const float* __restrict__ bhh,      // (192)
    const float* __restrict__ decW1,    // (128,128)
    const float* __restrict__ decB1,    // (128)
    const float* __restrict__ lhW1,     // (128,128)
    const float* __restrict__ lhB1,     // (128)
    const float* __restrict__ lhW2,     // (128,1)
    const float* __restrict__ lhB2,     // (1)
    float* __restrict__ a_dec,          // (B*T,128): row b*T+t
    float* __restrict__ out_lh,         // (B,T)
    float* __restrict__ out_z) {        // (B,T,64)
  __shared__ float s_h[BB * DH];        // hidden state (32x64)
  __shared__ float s_feat[BB * 128];    // [z | h_new]  (32x128)
  __shared__ float s_buf1[BB * 128];    // g -> ggate(r,u) -> a_lh
  __shared__ float s_buf2[BB * 128];    // logits -> [gin | ghn]

  const int tid = threadIdx.x;
  const int wave = tid >> 5;
  const int lane = tid & 31;

  for (int i = tid; i < BB * DH; i += 256) s_h[i] = 0.0f;
  __syncthreads();

  const float* latW1h = latW1 + DMODEL * DMLP;  // rows 768..831

  for (int t = 0; t < TT; ++t) {
    // ---- Stage 1: g = GELU(U_t + h @ latW1h + b1)      (32x128) ----
    for (int tile = wave; tile < 16; tile += 8) {
      const int m0 = (tile >> 3) * 16, n0 = (tile & 7) * 16;
      // init C = U[(b*T+t), n] + b1[n] ; rows stride T*128 in U
      v8f c = loadC(U + ((size_t)m0 * TT + t) * DMLP + n0, TT * DMLP);
      {
        const float bv = latB1[n0 + (lane & 15)];
        for (int r = 0; r < 8; ++r) c[r] += bv;
      }
      c = tileGemm<false>(s_h + m0 * DH, DH, latW1h + n0, DMLP, DH, c);
      c = geluC(c);
      storeC(s_buf1 + m0 * 128 + n0, 128, c);
    }
    __syncthreads();

    // ---- Stage 2: logits = g @ latW2 + b2              (32x64) ----
    {
      const int m0 = (wave >> 2) * 16, n0 = (wave & 3) * 16;
      v8f c = biasC(latB2, n0);
      c = tileGemm<false>(s_buf1 + m0 * 128, 128, latW2 + n0, DZ, DMLP, c);
      storeC(s_buf2 + m0 * DZ + n0, DZ, c);
    }
    __syncthreads();

    // ---- Stage 3: softmax / unimix / straight-through one-hot ----
    {
      const int b = tid >> 3, cat = tid & 7;
      const float* l = s_buf2 + b * DZ + cat * NCLS;
      float m = l[0];
      for (int c = 1; c < NCLS; ++c) m = fmaxf(m, l[c]);
      float e[NCLS], s = 0.0f;
      for (int c = 0; c < NCLS; ++c) { e[c] = expf(l[c] - m); s += e[c]; }
      const float inv = 1.0f / s;
      float p[NCLS];
      float best = -1.0f; int bi = 0;
      for (int c = 0; c < NCLS; ++c) {
        p[c] = 0.99f * (e[c] * inv) + 0.00125f;  // unimix 0.01, 8 classes
        if (p[c] > best) { best = p[c]; bi = c; }
      }
      float* zf = s_feat + b * 128 + cat * NCLS;
      float* zo = out_z + ((size_t)b * TT + t) * DZ + cat * NCLS;
      for (int c = 0; c < NCLS; ++c) {
        const float hd = (c == bi) ? 1.0f : 0.0f;
        const float zv = (hd + p[c]) - p[c];  // straight-through (fwd == hard)
        zf[c] = zv;
        zo[c] = zv;
      }
    }
    __syncthreads();

    // ---- Stage 4: GRU gate GEMMs ----
    // tiles 0..15 : ggate(32x128) = z@Wih[:128].T + h@Whh[:128].T + bih+bhh
    // tiles 16..23: gin (32x64)   = z@Wih[128:].T + bih[128:]
    // tiles 24..31: ghn (32x64)   = h@Whh[128:].T + bhh[128:]
    for (int tile = wave; tile < 32; tile += 8) {
      if (tile < 16) {
        const int m0 = (tile >> 3) * 16, n0 = (tile & 7) * 16;
        v8f c;
        {
          const int col = n0 + (lane & 15);
          const float bv = bih[col] + bhh[col];
          for (int r = 0; r < 8; ++r) c[r] = bv;
        }
        c = tileGemm<true>(s_feat + m0 * 128, 128, Wih + n0 * DH, DH, DH, c);
        c = tileGemm<true>(s_h + m0 * DH, DH, Whh + n0 * DH, DH, DH, c);
        storeC(s_buf1 + m0 * 128 + n0, 128, c);
      } else if (tile < 24) {
        const int q = tile - 16;
        const int m0 = (q >> 2) * 16, nt = (q & 3);
        const int n0 = 128 + nt * 16;
        v8f c = biasC(bih, n0);
        c = tileGemm<true>(s_feat + m0 * 128, 128, Wih + n0 * DH, DH, DH, c);
        storeC(s_buf2 + m0 * 128 + nt * 16, 128, c);  // gin in cols 0..63
      } else {
        const int q = tile - 24;
        const int m0 = (q >> 2) * 16, nt = (q & 3);
        const int n0 = 128 + nt * 16;
        v8f c = biasC(bhh, n0);
        c = tileGemm<true>(s_h + m0 * DH, DH, Whh + n0 * DH, DH, DH, c);
        storeC(s_buf2 + m0 * 128 + 64 + nt * 16, 128, c);  // ghn in cols 64..127
      }
    }
    __syncthreads();

    // ---- Stage 5: GRU elementwise -> h_new ----
    for (int idx = tid; idx < BB * DH; idx += 256) {
      const int b = idx >> 6, j = idx & 63;
      const float r = sigmoidf(s_buf1[b * 128 + j]);
      const float u = sigmoidf(s_buf1[b * 128 + 64 + j]);
      const float gin = s_buf2[b * 128 + j];
      const float ghn = s_buf2[b * 128 + 64 + j];
      const float n = tanhf(gin + r * ghn);
      const float hn = (1.0f - u) * n + u * s_h[idx];
      s_h[idx] = hn;                    // per-element in-place (safe)
      s_feat[b * 128 + 64 + j] = hn;    // feat = [z | h_new]
    }
    __syncthreads();

    // ---- Stage 6/7: a_dec = GELU(feat@decW1+b), a_lh = GELU(feat@lhW1+b) ----
    for (int tile = wave; tile < 32; tile += 8) {
      const int q = tile & 15;
      const int m0 = (q >> 3) * 16, n0 = (q & 7) * 16;
      if (tile < 16) {
        v8f c = biasC(decB1, n0);
        c = tileGemm<false>(s_feat + m0 * 128, 128, decW1 + n0, DMLP, DMLP, c);
        c = geluC(c);
        // store to global scratch, rows (b*T + t)
        storeC(a_dec + ((size_t)m0 * TT + t) * DMLP + n0, TT * DMLP, c);
      } else {
        v8f c = biasC(lhB1, n0);
        c = tileGemm<false>(s_feat + m0 * 128, 128, lhW1 + n0, DMLP, DMLP, c);
        c = geluC(c);
        storeC(s_buf1 + m0 * 128 + n0, 128, c);
      }
    }
    __syncthreads();

    // ---- Stage 8: lh[b] = a_lh[b,:] . lhW2 + lhB2 ----
    if (tid < BB) {
      float acc = lhB2[0];
      for (int k = 0; k < DMLP; ++k) acc += s_buf1[tid * 128 + k] * lhW2[k];
      out_lh[(size_t)tid * TT + t] = acc;
    }
    __syncthreads();  // buf1 free for next iteration's stage 1
  }
}

// ---------------------------------------------------------------------------
extern "C" void kernel_launch(void* const* d_in, const int* in_sizes, int n_in,
                              void* d_out, int out_size, void* d_ws,
                              size_t ws_size, hipStream_t stream) {
  const float* r1    = (const float*)d_in[0];
  const float* latW1 = (const float*)d_in[1];
  const float* latB1 = (const float*)d_in[2];
  const float* latW2 = (const float*)d_in[3];
  const float* latB2 = (const float*)d_in[4];
  const float* Wih   = (const float*)d_in[5];
  const float* bih   = (const float*)d_in[6];
  const float* Whh   = (const float*)d_in[7];
  const float* bhh   = (const float*)d_in[8];
  const float* decW1 = (const float*)d_in[9];
  const float* decB1 = (const float*)d_in[10];
  const float* decW2 = (const float*)d_in[11];
  const float* decB2 = (const float*)d_in[12];
  const float* lhW1  = (const float*)d_in[13];
  const float* lhB1  = (const float*)d_in[14];
  const float* lhW2  = (const float*)d_in[15];
  const float* lhB2  = (const float*)d_in[16];

  float* out   = (float*)d_out;
  float* r2    = out;                                   // (B,T,768)
  float* o_lh  = out + (size_t)BB * TT * DMODEL;        // (B,T)
  float* o_z   = o_lh + (size_t)BB * TT;                // (B,T,64)

  float* ws    = (float*)d_ws;
  float* U     = ws;                                    // (B*T,128)
  float* a_dec = ws + (size_t)BB * TT * DMLP;           // (B*T,128)

  // Phase 1: precompute h-independent part of latent MLP (parallel WMMA)
  phase1_kernel<<<1024, 256, 0, stream>>>(r1, latW1, U);

  // Phase 2: sequential scan (one persistent WGP-resident workgroup)
  seq_kernel<<<1, 256, 0, stream>>>(U, latW1, latB1, latW2, latB2, Wih, bih,
                                    Whh, bhh, decW1, decB1, lhW1, lhB1, lhW2,
                                    lhB2, a_dec, o_lh, o_z);

  // Phase 3: deferred decoder second layer (parallel WMMA)
  phase3_kernel<<<6144, 256, 0, stream>>>(a_dec, decW2, decB2, r2);
}